// HybridQuantumMLP_3255585210673
// MI455X (gfx1250) — compile-verified
//
#include <hip/hip_runtime.h>
#include <math.h>

#define NQ 6
#define NL 6
#define WPB 16            // waves (items) per block
#define TWO_PI_F 6.2831853f
#define PROB_STRIDE 66    // padded LDS row stride (floats): 8B-aligned, conflict-free

typedef float v2f __attribute__((ext_vector_type(2)));
typedef float v8f __attribute__((ext_vector_type(8)));

struct QState { float r0, i0, r1, i1; };   // lane holds amplitudes k=lane (slot0), k=lane+32 (slot1)

__device__ __forceinline__ float sxor(float v, int mask) { return __shfl_xor(v, mask, 32); }

// Broadcast lane `l`'s value to the whole wave as a scalar (v_readlane_b32 -> SGPR).
__device__ __forceinline__ float rdlane(float v, int l) {
    return __int_as_float(__builtin_amdgcn_readlane(__float_as_int(v), l));
}

// RY on amplitude bit `bit` (bit = 5 - wire). c=cos(th/2), s=sin(th/2).
__device__ __forceinline__ void ry_gate(QState& q, int bit, float c, float s, int lane) {
    if (bit == 5) { // in-lane slot pair
        float nr0 = c*q.r0 - s*q.r1, ni0 = c*q.i0 - s*q.i1;
        float nr1 = s*q.r0 + c*q.r1, ni1 = s*q.i0 + c*q.i1;
        q.r0 = nr0; q.i0 = ni0; q.r1 = nr1; q.i1 = ni1;
    } else {
        int m = 1 << bit;
        float pr0 = sxor(q.r0, m), pi0 = sxor(q.i0, m);
        float pr1 = sxor(q.r1, m), pi1 = sxor(q.i1, m);
        float sg = ((lane >> bit) & 1) ? s : -s;  // lo: c*a0 - s*a1 ; hi: s*a0 + c*a1
        q.r0 = fmaf(sg, pr0, c*q.r0); q.i0 = fmaf(sg, pi0, c*q.i0);
        q.r1 = fmaf(sg, pr1, c*q.r1); q.i1 = fmaf(sg, pi1, c*q.i1);
    }
}

// RZ: amplitude *= exp(i*sigma*th/2), sigma = -1 if bit==0 else +1.
__device__ __forceinline__ void rz_gate(QState& q, int bit, float c, float s, int lane) {
    if (bit == 5) { // slot0 sigma=-1, slot1 sigma=+1
        float nr0 = c*q.r0 + s*q.i0, ni0 = c*q.i0 - s*q.r0;
        float nr1 = c*q.r1 - s*q.i1, ni1 = c*q.i1 + s*q.r1;
        q.r0 = nr0; q.i0 = ni0; q.r1 = nr1; q.i1 = ni1;
    } else {
        float sg = ((lane >> bit) & 1) ? s : -s;
        float nr0 = c*q.r0 - sg*q.i0, ni0 = c*q.i0 + sg*q.r0;
        float nr1 = c*q.r1 - sg*q.i1, ni1 = c*q.i1 + sg*q.r1;
        q.r0 = nr0; q.i0 = ni0; q.r1 = nr1; q.i1 = ni1;
    }
}

// CNOT, ctrl bit 5 (wire 0), tgt lane bit bt: slot1 permutes across lanes.
__device__ __forceinline__ void cnot_c5(QState& q, int bt) {
    int m = 1 << bt;
    q.r1 = sxor(q.r1, m); q.i1 = sxor(q.i1, m);
}
// CNOT, both bits are lane bits: lanes with ctrl bit set swap with lane^(1<<bt).
__device__ __forceinline__ void cnot_lane(QState& q, int bc, int bt, int lane) {
    int m = 1 << bt;
    float pr0 = sxor(q.r0, m), pi0 = sxor(q.i0, m);
    float pr1 = sxor(q.r1, m), pi1 = sxor(q.i1, m);
    bool f = (lane >> bc) & 1;
    q.r0 = f ? pr0 : q.r0; q.i0 = f ? pi0 : q.i0;
    q.r1 = f ? pr1 : q.r1; q.i1 = f ? pi1 : q.i1;
}
// CNOT, tgt bit 5: lanes with ctrl lane-bit set swap slot0 <-> slot1.
__device__ __forceinline__ void cnot_t5(QState& q, int bc, int lane) {
    bool f = (lane >> bc) & 1;
    float tr = q.r0, ti = q.i0;
    q.r0 = f ? q.r1 : q.r0; q.i0 = f ? q.i1 : q.i0;
    q.r1 = f ? tr  : q.r1;  q.i1 = f ? ti  : q.i1;
}

__global__ __launch_bounds__(32 * WPB)
void qsim_wmma_kernel(const float* __restrict__ x, const float* __restrict__ w,
                      float* __restrict__ out, int nitems) {
    __shared__ float probs[WPB * PROB_STRIDE];

    const int lane = threadIdx.x & 31;
    const int wave = threadIdx.x >> 5;
    const int item = blockIdx.x * WPB + wave;
    const bool active = (item < nitems);   // wave-uniform

    QState q;
    q.r0 = (lane == 0) ? 1.0f : 0.0f;
    q.i0 = 0.0f; q.r1 = 0.0f; q.i1 = 0.0f;

    if (active) {
        // ---- encode ----
        float x0 = x[item * 3 + 0], x1 = x[item * 3 + 1], x2 = x[item * 3 + 2];
        float th = acosf(fminf(fmaxf(x2, -1.0f), 1.0f));
        float ph = fmodf(atan2f(x1, x0) + TWO_PI_F, TWO_PI_F);
        // one lane-parallel sincos: lane0 -> theta/2, lane1 -> phi/2
        float ev = 0.5f * ((lane & 1) ? ph : th);
        float es, ec;
        __sincosf(ev, &es, &ec);
        float ct = rdlane(ec, 0), st = rdlane(es, 0);
        float cp = rdlane(ec, 1), sp = rdlane(es, 1);
        #pragma unroll
        for (int wire = 0; wire < NQ; ++wire) {
            ry_gate(q, 5 - wire, ct, st, lane);
            rz_gate(q, 5 - wire, cp, sp, lane);
        }
        // ---- variational layers ----
        const float* wp = w + (size_t)item * (NL * 2 * NQ);
        const int widx = (lane < 12) ? lane : 11;   // lanes 0..11 carry this layer's 12 angles
        for (int l = 0; l < NL; ++l) {
            // single wave-wide sincos for all 12 gate angles of this layer
            float ang = wp[l * 12 + widx];
            float hs, hc;
            __sincosf(0.5f * ang, &hs, &hc);
            #pragma unroll
            for (int i = 0; i < NQ; ++i) {
                float cz = rdlane(hc, 2 * i),     sz = rdlane(hs, 2 * i);
                rz_gate(q, 5 - i, cz, sz, lane);
                float cy = rdlane(hc, 2 * i + 1), sy = rdlane(hs, 2 * i + 1);
                ry_gate(q, 5 - i, cy, sy, lane);
            }
            cnot_c5(q, 4);            // CNOT(0,1): ctrl bit5, tgt bit4
            cnot_lane(q, 4, 3, lane); // CNOT(1,2)
            cnot_lane(q, 3, 2, lane); // CNOT(2,3)
            cnot_lane(q, 2, 1, lane); // CNOT(3,4)
            cnot_lane(q, 1, 0, lane); // CNOT(4,5)
            cnot_t5(q, 0, lane);      // CNOT(5,0): ctrl bit0, tgt bit5
        }
    }

    // probabilities -> LDS (inactive waves write zeros so WMMA rows are clean)
    float plo = active ? (q.r0 * q.r0 + q.i0 * q.i0) : 0.0f;
    float phi = active ? (q.r1 * q.r1 + q.i1 * q.i1) : 0.0f;
    probs[wave * PROB_STRIDE + lane]      = plo;
    probs[wave * PROB_STRIDE + 32 + lane] = phi;
    __syncthreads();

    // ---- Pauli-Z contraction: exp[16 items][16 cols] = probs[16x64] * signs[64x16] ----
    // 16 chained V_WMMA_F32_16X16X4_F32 (K=4 each), sign matrix generated in registers.
    if (wave == 0) {
        const int half = lane >> 4;   // 0: K pair {0,1}; 1: K pair {2,3}
        const int m    = lane & 15;   // A row (item) / B,D column
        v8f acc = {0.f, 0.f, 0.f, 0.f, 0.f, 0.f, 0.f, 0.f};
        #pragma unroll
        for (int c = 0; c < 16; ++c) {
            int kb = 4 * c + 2 * half;
            v2f a, b;
            a.x = probs[m * PROB_STRIDE + kb];
            a.y = probs[m * PROB_STRIDE + kb + 1];
            // signs[k][q] = 1 - 2*((k >> (5-q)) & 1), columns 6..15 are zero
            b.x = (m < NQ) ? (1.0f - 2.0f * (float)((kb >> (5 - m)) & 1)) : 0.0f;
            b.y = (m < NQ) ? (1.0f - 2.0f * (float)(((kb + 1) >> (5 - m)) & 1)) : 0.0f;
            acc = __builtin_amdgcn_wmma_f32_16x16x4_f32(
                /*neg_a=*/false, a, /*neg_b=*/false, b,
                /*c_mod=*/(short)0, acc, /*reuse_a=*/false, /*reuse_b=*/false);
        }
        // D layout: VGPR v -> row (v + 8*half), col = m
        const int base_item = blockIdx.x * WPB;
        #pragma unroll
        for (int v = 0; v < 8; ++v) {
            int it = base_item + v + 8 * half;
            if (m < NQ && it < nitems) {
                float val = acc[v];
                out[it * NQ + m] = 1.0f / (1.0f + __expf(-val));
            }
        }
    }
}

extern "C" void kernel_launch(void* const* d_in, const int* in_sizes, int n_in,
                              void* d_out, int out_size, void* d_ws, size_t ws_size,
                              hipStream_t stream) {
    (void)n_in; (void)out_size; (void)d_ws; (void)ws_size;
    const float* x = (const float*)d_in[0];     // (B, 3) fp32
    const float* w = (const float*)d_in[1];     // (B, 72) fp32
    float* out = (float*)d_out;                 // (B, 6) fp32
    int b = in_sizes[0] / 3;
    int blocks = (b + WPB - 1) / WPB;
    qsim_wmma_kernel<<<blocks, 32 * WPB, 0, stream>>>(x, w, out, b);
}